// PointConvDensity_15006615734069
// MI455X (gfx1250) — compile-verified
//
#include <hip/hip_runtime.h>
#include <math.h>

// ---------------------------------------------------------------------------
// PointConvDensity on gfx1250 (MI455X, wave32).
//
// Decomposition: feat channels 0..125 (+127 ctr, +bias) are s-invariant ->
// base[b,n,o] via V_WMMA_F32_16X16X4_F32 (16384x128x128 fp32 GEMM).
// Only u = nbr-2*ctr (ch 126) and v = 1/(|d|+1e-8) (ch 128) depend on s:
//   h = base + W126[o]*u + W128[o]*v
// BN stats computed analytically from per-row sums; max_s commutes with the
// monotone affine+relu (use min_s when scale<0).
// ---------------------------------------------------------------------------

typedef __attribute__((ext_vector_type(2))) float v2f;
typedef __attribute__((ext_vector_type(8))) float v8f;

constexpr int Bb  = 8;
constexpr int Nn  = 2048;
constexpr int Ss  = 32;
constexpr int OUT = 128;
constexpr int Dd  = 126;   // point channels
constexpr int CIN = 129;
constexpr int ROWS = Bb * Nn;      // 16384
constexpr int KK  = 128;           // padded GEMM K

// workspace layout (float offsets); total ~12.98 MB
constexpr int WT_OFF    = 0;                       // 128*128 transposed W (K-major)
constexpr int W126_OFF  = WT_OFF   + KK * OUT;     // 128
constexpr int W128_OFF  = W126_OFF + OUT;          // 128
constexpr int SCALE_OFF = W128_OFF + OUT;          // 128
constexpr int SHIFT_OFF = SCALE_OFF + OUT;         // 128
constexpr int TS_OFF    = SHIFT_OFF + OUT;         // 8 (5 used: Tsu,Tsv,Tsuu,Tsvv,Tsuv)
constexpr int SU_OFF    = TS_OFF  + 8;             // 16384
constexpr int SV_OFF    = SU_OFF  + ROWS;
constexpr int SUU_OFF   = SV_OFF  + ROWS;
constexpr int SVV_OFF   = SUU_OFF + ROWS;
constexpr int SUV_OFF   = SVV_OFF + ROWS;
constexpr int U_OFF     = SUV_OFF + ROWS;          // 524288
constexpr int V_OFF     = U_OFF   + ROWS * Ss;     // 524288
constexpr int BASE_OFF  = V_OFF   + ROWS * Ss;     // 16384*128

// ---- kernel 0: build K-major padded weight matrix + split-out W126/W128 ----
__global__ void k_prep(const float* __restrict__ W, const float* __restrict__ bias,
                       float* __restrict__ ws) {
  int t = blockIdx.x * blockDim.x + threadIdx.x;   // 16384 threads
  int k = t >> 7, o = t & 127;
  float wt;
  if (k < Dd)        wt = W[o * CIN + k];          // pts-ctr channels
  else if (k == 126) wt = W[o * CIN + 127];        // ctr channel
  else               wt = bias[o];                 // bias lane (feat = 1)
  ws[WT_OFF + k * OUT + o] = wt;
  if (k == 0) {
    ws[W126_OFF + o] = W[o * CIN + 126];
    ws[W128_OFF + o] = W[o * CIN + 128];
  }
}

// ---- kernel S: per-row neighbor stats + u/v streams -----------------------
__global__ void k_stats(const float* __restrict__ xyz, const int* __restrict__ idx,
                        float* __restrict__ ws) {
  int row = blockIdx.x * blockDim.x + threadIdx.x;
  if (row >= ROWS) return;
  int b = row >> 11;
  float ctr = xyz[row];
  const int* ip = idx + row * Ss;
  float su = 0.f, sv = 0.f, suu = 0.f, svv = 0.f, suv = 0.f;
  for (int s = 0; s < Ss; ++s) {
    float nbr = xyz[b * Nn + ip[s]];
    float d = nbr - ctr;
    float u = d - ctr;                       // diff - center
    float v = 1.0f / (fabsf(d) + 1e-8f);     // density (1-D -> |d|)
    ws[U_OFF + row * Ss + s] = u;
    ws[V_OFF + row * Ss + s] = v;
    su += u; sv += v; suu += u * u; svv += v * v; suv += u * v;
  }
  ws[SU_OFF + row] = su;  ws[SV_OFF + row] = sv;
  ws[SUU_OFF + row] = suu; ws[SVV_OFF + row] = svv; ws[SUV_OFF + row] = suv;
}

// ---- kernel R: reduce the 5 per-row sums to global scalars ----------------
__global__ void __launch_bounds__(256) k_reduce5(float* __restrict__ ws) {
  __shared__ float sm[5][256];
  int tid = threadIdx.x;
  float a0 = 0.f, a1 = 0.f, a2 = 0.f, a3 = 0.f, a4 = 0.f;
  for (int r = tid; r < ROWS; r += 256) {
    a0 += ws[SU_OFF + r];  a1 += ws[SV_OFF + r];  a2 += ws[SUU_OFF + r];
    a3 += ws[SVV_OFF + r]; a4 += ws[SUV_OFF + r];
  }
  sm[0][tid] = a0; sm[1][tid] = a1; sm[2][tid] = a2; sm[3][tid] = a3; sm[4][tid] = a4;
  __syncthreads();
  for (int off = 128; off > 0; off >>= 1) {
    if (tid < off)
      for (int j = 0; j < 5; ++j) sm[j][tid] += sm[j][tid + off];
    __syncthreads();
  }
  if (tid < 5) ws[TS_OFF + tid] = sm[tid][0];
}

// ---- kernel 1: base = Afeat(16384x128) x Wt(128x128) via WMMA fp32 --------
// One wave32 per (16-row M-tile) x (two 16-col N-tiles): 2 accumulators share
// each A operand -> 2 WMMAs per A load pair, fully unrolled, branchless A
// (EXEC never diverges; WMMA requires EXEC all-1s).
__global__ void __launch_bounds__(32) k_base(const float* __restrict__ xyz,
                                             const float* __restrict__ pts,
                                             float* __restrict__ ws) {
  int tile = blockIdx.x;                   // 1024 M-tiles x 4 N-pairs = 4096
  int tm = tile >> 2, tp = tile & 3;
  int lane = threadIdx.x;
  int m = lane & 15, g = lane >> 4;        // lane-group halves hold K = 2g+j
  int row = tm * 16 + m;
  int b = row >> 11, n = row & 2047;
  float ctr = xyz[row];
  int o0 = tp * 32 + m;                    // first N-tile column for this lane
  int o1 = o0 + 16;                        // second N-tile column
  const float* __restrict__ wt = ws + WT_OFF;
  const float* __restrict__ pbase = pts + (size_t)b * Dd * Nn + n;

  v8f acc0 = {};
  v8f acc1 = {};
#pragma unroll
  for (int k0 = 0; k0 < KK; k0 += 4) {
    int k = k0 + 2 * g;                    // even; k in [0,126]
    // Branchless A: clamp channel for the load (always in-bounds), then
    // select the padded-channel value. k==126 -> ctr; k+1==127 -> 1.0 (bias).
    int kc0 = k     < Dd ? k     : Dd - 1;
    int kc1 = k + 1 < Dd ? k + 1 : Dd - 1;
    float p0 = pbase[kc0 * Nn] - ctr;
    float p1 = pbase[kc1 * Nn] - ctr;
    v2f a;
    a.x = (k     < Dd) ? p0 : ctr;
    a.y = (k + 1 < Dd) ? p1 : 1.0f;
    v2f b0, b1;
    b0.x = wt[k * OUT + o0];               // B 4x16: VGPR j = K (2g+j), N = lane&15
    b0.y = wt[(k + 1) * OUT + o0];
    b1.x = wt[k * OUT + o1];
    b1.y = wt[(k + 1) * OUT + o1];
    acc0 = __builtin_amdgcn_wmma_f32_16x16x4_f32(
        false, a, false, b0, (short)0, acc0, false, false);
    acc1 = __builtin_amdgcn_wmma_f32_16x16x4_f32(
        false, a, false, b1, (short)0, acc1, false, false);
  }
  float* __restrict__ base = ws + BASE_OFF;
#pragma unroll
  for (int j = 0; j < 8; ++j) {            // C/D: VGPR j -> M = j + 8g, N = lane&15
    base[(tm * 16 + 8 * g + j) * OUT + o0] = acc0[j];
    base[(tm * 16 + 8 * g + j) * OUT + o1] = acc1[j];
  }
}

// ---- kernel 2: per-channel BN scale/shift from analytic moments -----------
__global__ void __launch_bounds__(256) k_chanstats(const float* __restrict__ gamma,
                                                   const float* __restrict__ beta,
                                                   float* __restrict__ ws) {
  int o = blockIdx.x, tid = threadIdx.x;
  __shared__ float sm[4][256];
  float sb = 0.f, sbb = 0.f, sbu = 0.f, sbv = 0.f;
  for (int r = tid; r < ROWS; r += 256) {
    float bse = ws[BASE_OFF + r * OUT + o];
    sb  += bse;
    sbb += bse * bse;
    sbu += bse * ws[SU_OFF + r];
    sbv += bse * ws[SV_OFF + r];
  }
  sm[0][tid] = sb; sm[1][tid] = sbb; sm[2][tid] = sbu; sm[3][tid] = sbv;
  __syncthreads();
  for (int off = 128; off > 0; off >>= 1) {
    if (tid < off)
      for (int j = 0; j < 4; ++j) sm[j][tid] += sm[j][tid + off];
    __syncthreads();
  }
  if (tid == 0) {
    float Sb = sm[0][0], Sbb = sm[1][0], Sbu = sm[2][0], Sbv = sm[3][0];
    float w1 = ws[W126_OFF + o], w2 = ws[W128_OFF + o];
    float Tsu = ws[TS_OFF], Tsv = ws[TS_OFF + 1], Tsuu = ws[TS_OFF + 2];
    float Tsvv = ws[TS_OFF + 3], Tsuv = ws[TS_OFF + 4];
    const float invN = 1.0f / (float)(ROWS * Ss);
    float sum_h  = (float)Ss * Sb + w1 * Tsu + w2 * Tsv;
    float sum_h2 = (float)Ss * Sbb + w1 * w1 * Tsuu + w2 * w2 * Tsvv
                 + 2.f * w1 * Sbu + 2.f * w2 * Sbv + 2.f * w1 * w2 * Tsuv;
    float mean = sum_h * invN;
    float var  = sum_h2 * invN - mean * mean;
    float rstd = rsqrtf(var + 1e-5f);
    float sc = gamma[o] * rstd;
    ws[SCALE_OFF + o] = sc;
    ws[SHIFT_OFF + o] = beta[o] - mean * sc;
  }
}

// ---- kernel 3: out[row,o] = relu(scale*(base + ext_s(w1*u+w2*v)) + shift) --
__global__ void __launch_bounds__(128) k_out(float* __restrict__ out,
                                             const float* __restrict__ ws) {
  int row = blockIdx.x;
  int o = threadIdx.x;
  __shared__ float su_[Ss], sv_[Ss];
  if (o < Ss) {
    su_[o] = ws[U_OFF + row * Ss + o];
    sv_[o] = ws[V_OFF + row * Ss + o];
  }
  __syncthreads();
  float w1 = ws[W126_OFF + o], w2 = ws[W128_OFF + o];
  float bse = ws[BASE_OFF + row * OUT + o];
  float tmax = -3.4e38f, tmin = 3.4e38f;
#pragma unroll
  for (int s = 0; s < Ss; ++s) {
    float t = fmaf(w1, su_[s], w2 * sv_[s]);
    tmax = fmaxf(tmax, t);
    tmin = fminf(tmin, t);
  }
  float sc = ws[SCALE_OFF + o];
  float h = bse + (sc >= 0.f ? tmax : tmin);   // max_s commutes with monotone map
  float r = fmaf(sc, h, ws[SHIFT_OFF + o]);
  out[row * OUT + o] = fmaxf(r, 0.f);
}

extern "C" void kernel_launch(void* const* d_in, const int* in_sizes, int n_in,
                              void* d_out, int out_size, void* d_ws, size_t ws_size,
                              hipStream_t stream) {
  const float* xyz   = (const float*)d_in[0];
  const float* pts   = (const float*)d_in[1];
  const int*   idx   = (const int*)  d_in[2];
  const float* W     = (const float*)d_in[3];
  const float* bias  = (const float*)d_in[4];
  const float* gamma = (const float*)d_in[5];
  const float* beta  = (const float*)d_in[6];
  float* ws  = (float*)d_ws;   // needs ~13 MB
  float* out = (float*)d_out;

  k_prep     <<<64,          256, 0, stream>>>(W, bias, ws);
  k_stats    <<<ROWS / 256,  256, 0, stream>>>(xyz, idx, ws);
  k_reduce5  <<<1,           256, 0, stream>>>(ws);
  k_base     <<<(ROWS / 16) * (OUT / 32), 32, 0, stream>>>(xyz, pts, ws);
  k_chanstats<<<OUT,         256, 0, stream>>>(gamma, beta, ws);
  k_out      <<<ROWS,        128, 0, stream>>>(out, ws);
}